// UserTower_43954695307908
// MI455X (gfx1250) — compile-verified
//
#include <hip/hip_runtime.h>
#include <hip/hip_bf16.h>

typedef __attribute__((ext_vector_type(2))) float v2f;
typedef __attribute__((ext_vector_type(8))) float v8f;

#define LN_EPS 1e-5f

// ---------------------------------------------------------------------------
// Kernel 1: per-user feature gather + mean pooling -> xbuf[B][192] (fp32)
// One wave (32 lanes) per user. All row gathers are 128B-coalesced.
// ---------------------------------------------------------------------------
__global__ __launch_bounds__(256) void gather_kernel(
    const int* __restrict__ user_input,
    const float* __restrict__ emb,            // [N,64]
    const int* __restrict__ ctry_idx,         // [N]
    const float* __restrict__ ctry_emb,       // [256,32]
    const int* __restrict__ dev_idx,          // [N]
    const float* __restrict__ dev_emb,        // [10000,32]
    const int* __restrict__ tags_idx,         // [N,20]
    const float* __restrict__ tags_emb,       // [50000,32]
    const int* __restrict__ hist_idx,         // [N,50]
    const float* __restrict__ hist_emb,       // [100000,32]
    float* __restrict__ xbuf,                 // [B,192]
    int B)
{
    const int wave = (blockIdx.x * blockDim.x + threadIdx.x) >> 5;
    const int lane = threadIdx.x & 31;
    if (wave >= B) return;

    const long idx = (long)user_input[wave];
    float* __restrict__ row = xbuf + (long)wave * 192;

    // [0,64): item embedding
    const float* e = emb + idx * 64;
    row[lane]      = e[lane];
    row[lane + 32] = e[lane + 32];

    // [64,96): country embedding
    const long c = (long)ctry_idx[idx];
    row[64 + lane] = ctry_emb[c * 32 + lane];

    // [96,128): device embedding
    const long dv = (long)dev_idx[idx];
    row[96 + lane] = dev_emb[dv * 32 + lane];

    // [128,160): tags mean-pool over 20
    {
        const int* ti = tags_idx + idx * 20;
        float acc = 0.f;
        #pragma unroll 4
        for (int w = 0; w < 20; ++w)
            acc += tags_emb[(long)ti[w] * 32 + lane];
        row[128 + lane] = acc * (1.f / 20.f);
    }
    // [160,192): history mean-pool over 50
    {
        const int* hi = hist_idx + idx * 50;
        float acc = 0.f;
        #pragma unroll 5
        for (int w = 0; w < 50; ++w)
            acc += hist_emb[(long)hi[w] * 32 + lane];
        row[160 + lane] = acc * (1.f / 50.f);
    }
}

// ---------------------------------------------------------------------------
// Kernel 2: y = relu(x @ fc_w^T + fc_b) ; layernorm(y) -> out[B][256]
// One block (256 thr = 8 waves) per 16-user M-tile. Each wave: 2 N-tiles of 16,
// K=192 in steps of 4 via v_wmma_f32_16x16x4_f32 (48 wmma per tile).
// ---------------------------------------------------------------------------
__global__ __launch_bounds__(256) void fc_ln_kernel(
    const float* __restrict__ xbuf,   // [B,192]
    const float* __restrict__ fc_w,   // [256,192] row-major (out-major)
    const float* __restrict__ fc_b,   // [256]
    const float* __restrict__ ln_g,   // [256]
    const float* __restrict__ ln_b,   // [256]
    float* __restrict__ out,          // [B,256]
    int B)
{
    __shared__ float xs[16][256];     // relu(fc) tile
    __shared__ float ps[16][16];      // LN partial sums
    __shared__ float ps2[16][16];     // LN partial sumsq
    __shared__ float mu[16];
    __shared__ float rs[16];

    const int u0   = blockIdx.x * 16;
    const int tid  = threadIdx.x;
    const int wave = tid >> 5;
    const int lane = tid & 31;

    // ---- A fragment addressing (f32 16x4 layout):
    // lanes 0-15: M=lane, K = k0+{0,1}; lanes 16-31: M=lane-16, K = k0+{2,3}
    const int m    = lane & 15;
    const int koff = (lane >> 4) * 2;
    int urow = u0 + m;
    if (urow >= B) urow = B - 1;           // clamp (B is a multiple of 16 anyway)
    const float* arow = xbuf + (long)urow * 192 + koff;

    // ---- B fragment addressing (f32 4x16 layout, mirror of A):
    // lanes 0-15: N=lane, K = k0+{0,1}; lanes 16-31: N=lane-16, K = k0+{2,3}
    // B[k][n] = fc_w[n0+n][k]  ->  contiguous float2 from fc_w row (n0+n).
    const int ncol = lane & 15;
    const int n0 = wave * 32;
    const int n1 = n0 + 16;
    const float* brow0 = fc_w + (long)(n0 + ncol) * 192 + koff;
    const float* brow1 = fc_w + (long)(n1 + ncol) * 192 + koff;

    v8f c0 = {};
    v8f c1 = {};

    #pragma unroll 4
    for (int k0 = 0; k0 < 192; k0 += 4) {
        v2f a  = *(const v2f*)(arow  + k0);
        v2f b0 = *(const v2f*)(brow0 + k0);
        v2f b1 = *(const v2f*)(brow1 + k0);
        c0 = __builtin_amdgcn_wmma_f32_16x16x4_f32(
                 false, a, false, b0, (short)0, c0, false, false);
        c1 = __builtin_amdgcn_wmma_f32_16x16x4_f32(
                 false, a, false, b1, (short)0, c1, false, false);
    }

    // ---- bias + relu into LDS tile using C/D layout:
    // element (vgpr i, lane l): M = i + 8*(l>=16), N = l&15
    const float bias0 = fc_b[n0 + ncol];
    const float bias1 = fc_b[n1 + ncol];
    const int mbase = (lane >> 4) * 8;
    #pragma unroll
    for (int i = 0; i < 8; ++i) {
        float v0 = c0[i] + bias0;
        float v1 = c1[i] + bias1;
        xs[mbase + i][n0 + ncol] = v0 > 0.f ? v0 : 0.f;
        xs[mbase + i][n1 + ncol] = v1 > 0.f ? v1 : 0.f;
    }
    __syncthreads();

    // ---- LayerNorm over 256 cols per row; 16 threads per row.
    const int r   = tid >> 4;
    const int seg = tid & 15;
    {
        float s = 0.f, s2 = 0.f;
        #pragma unroll
        for (int i = 0; i < 16; ++i) {
            float v = xs[r][seg + 16 * i];   // team covers consecutive cols
            s  += v;
            s2 += v * v;
        }
        ps[r][seg]  = s;
        ps2[r][seg] = s2;
    }
    __syncthreads();
    if (tid < 16) {
        float s = 0.f, s2 = 0.f;
        #pragma unroll
        for (int i = 0; i < 16; ++i) { s += ps[tid][i]; s2 += ps2[tid][i]; }
        float mean = s * (1.f / 256.f);
        float var  = s2 * (1.f / 256.f) - mean * mean;
        mu[tid] = mean;
        rs[tid] = rsqrtf(var + LN_EPS);
    }
    __syncthreads();

    if (u0 + r < B) {
        const float mean = mu[r];
        const float rstd = rs[r];
        float* orow = out + (long)(u0 + r) * 256;
        #pragma unroll
        for (int i = 0; i < 16; ++i) {
            int c = seg + 16 * i;            // 16-thread team -> 64B coalesced
            orow[c] = (xs[r][c] - mean) * rstd * ln_g[c] + ln_b[c];
        }
    }
}

// ---------------------------------------------------------------------------
extern "C" void kernel_launch(void* const* d_in, const int* in_sizes, int n_in,
                              void* d_out, int out_size, void* d_ws, size_t ws_size,
                              hipStream_t stream) {
    const int*   user_input = (const int*)  d_in[0];
    const float* emb        = (const float*)d_in[1];
    const int*   ctry_idx   = (const int*)  d_in[2];
    const float* ctry_emb   = (const float*)d_in[3];
    const int*   dev_idx    = (const int*)  d_in[4];
    const float* dev_emb    = (const float*)d_in[5];
    const int*   tags_idx   = (const int*)  d_in[6];
    const float* tags_emb   = (const float*)d_in[7];
    const int*   hist_idx   = (const int*)  d_in[8];
    const float* hist_emb   = (const float*)d_in[9];
    const float* fc_w       = (const float*)d_in[10];
    const float* fc_b       = (const float*)d_in[11];
    const float* ln_g       = (const float*)d_in[12];
    const float* ln_b       = (const float*)d_in[13];

    const int B = in_sizes[0];
    float* xbuf = (float*)d_ws;            // B*192 floats (12 MB for B=16384)
    float* outf = (float*)d_out;

    // Kernel 1: one wave per user, 8 waves per block.
    const int waves_per_block = 8;
    const int g1 = (B + waves_per_block - 1) / waves_per_block;
    gather_kernel<<<g1, 32 * waves_per_block, 0, stream>>>(
        user_input, emb, ctry_idx, ctry_emb, dev_idx, dev_emb,
        tags_idx, tags_emb, hist_idx, hist_emb, xbuf, B);

    // Kernel 2: one block per 16-user tile.
    const int g2 = (B + 15) / 16;
    fc_ln_kernel<<<g2, 256, 0, stream>>>(xbuf, fc_w, fc_b, ln_g, ln_b, outf, B);
}